// AddRandomWalkEdge_16896401342869
// MI455X (gfx1250) — compile-verified
//
#include <hip/hip_runtime.h>
#include <stdint.h>

// ---------------------------------------------------------------------------
// AddRandomWalkEdge for MI455X (gfx1250, wave32).
// Integer gather workload: WMMA does not apply (no dense GEMM anywhere).
// CDNA5-specific paths used: async global<->LDS DMA (ASYNCcnt) for the bulk
// prefix copies, split wait counters, wave32-sized blocks.
// ---------------------------------------------------------------------------

#define WALK_LEN 8
#define WALKS_PER_NODE 4

// ---- fills ----------------------------------------------------------------
__global__ void zero_i32_kernel(int* __restrict__ p, int n) {
  int i = blockIdx.x * blockDim.x + threadIdx.x;
  int n4 = n >> 2;
  if (i < n4) ((int4*)p)[i] = make_int4(0, 0, 0, 0);
  int t = (n4 << 2) + i;
  if (t < n) p[t] = 0;
}

__global__ void fill_ones_f32_kernel(float* __restrict__ p, int n) {
  int i = blockIdx.x * blockDim.x + threadIdx.x;
  int n4 = n >> 2;
  if (i < n4) ((float4*)p)[i] = make_float4(1.f, 1.f, 1.f, 1.f);
  int t = (n4 << 2) + i;
  if (t < n) p[t] = 1.f;
}

// ---- degree histogram (int4 vectorized) -----------------------------------
__global__ void hist_kernel(const int* __restrict__ row, int* __restrict__ deg,
                            int nE) {
  int i = blockIdx.x * blockDim.x + threadIdx.x;
  int stride = gridDim.x * blockDim.x;
  int n4 = nE >> 2;
  for (int j = i; j < n4; j += stride) {
    int4 r = ((const int4*)row)[j];
    atomicAdd(&deg[r.x], 1);
    atomicAdd(&deg[r.y], 1);
    atomicAdd(&deg[r.z], 1);
    atomicAdd(&deg[r.w], 1);
  }
  for (int j = (n4 << 2) + i; j < nE; j += stride) atomicAdd(&deg[row[j]], 1);
}

// ---- exclusive scan deg -> rowptr (+ packed (ptr,deg) pairs for the walk) -
#define SCAN_T 1024
__global__ void scan_kernel(const int* __restrict__ deg, int* __restrict__ rowptr,
                            int2* __restrict__ pd, int n) {
  __shared__ int ssum[SCAN_T];
  const int t = threadIdx.x;
  const int items = (n + SCAN_T - 1) / SCAN_T;
  const int base = t * items;
  int s = 0;
  for (int i = 0; i < items; ++i) {
    int idx = base + i;
    if (idx < n) s += deg[idx];
  }
  ssum[t] = s;
  __syncthreads();
  for (int off = 1; off < SCAN_T; off <<= 1) {
    int v = (t >= off) ? ssum[t - off] : 0;
    __syncthreads();
    ssum[t] += v;
    __syncthreads();
  }
  int run = ssum[t] - s;  // exclusive base
  for (int i = 0; i < items; ++i) {
    int idx = base + i;
    if (idx < n) {
      int d = deg[idx];
      rowptr[idx] = run;
      pd[idx] = make_int2(run, d);  // one b64 gather feeds the walk step
      run += d;
    }
  }
  if (t == SCAN_T - 1) rowptr[n] = ssum[t];
}

// ---- scatter original edge ids into CSR slots (order fixed up later) ------
__global__ void scatter_kernel(const int* __restrict__ row,
                               const int* __restrict__ rowptr,
                               int* __restrict__ cnt, int* __restrict__ idx_s,
                               int nE) {
  int i = blockIdx.x * blockDim.x + threadIdx.x;
  int stride = gridDim.x * blockDim.x;
  int n4 = nE >> 2;
  for (int j = i; j < n4; j += stride) {
    int4 r = ((const int4*)row)[j];
    int e = j << 2;
    idx_s[rowptr[r.x] + atomicAdd(&cnt[r.x], 1)] = e + 0;
    idx_s[rowptr[r.y] + atomicAdd(&cnt[r.y], 1)] = e + 1;
    idx_s[rowptr[r.z] + atomicAdd(&cnt[r.z], 1)] = e + 2;
    idx_s[rowptr[r.w] + atomicAdd(&cnt[r.w], 1)] = e + 3;
  }
  for (int j = (n4 << 2) + i; j < nE; j += stride) {
    int r = row[j];
    idx_s[rowptr[r] + atomicAdd(&cnt[r], 1)] = j;
  }
}

// ---- per-node insertion sort (restores stable argsort order) + col gather -
#define SEG_LOCAL 96
__global__ void segsort_kernel(const int* __restrict__ col,
                               const int* __restrict__ rowptr,
                               const int* __restrict__ deg,
                               int* __restrict__ idx_s, int* __restrict__ col_s,
                               int nN) {
  int v = blockIdx.x * blockDim.x + threadIdx.x;
  if (v >= nN) return;
  const int lo = rowptr[v];
  const int n = deg[v];
  if (n <= SEG_LOCAL) {
    int buf[SEG_LOCAL];
    for (int i = 0; i < n; ++i) buf[i] = idx_s[lo + i];
    for (int i = 1; i < n; ++i) {
      int key = buf[i];
      int j = i - 1;
      while (j >= 0 && buf[j] > key) { buf[j + 1] = buf[j]; --j; }
      buf[j + 1] = key;
    }
    for (int i = 0; i < n; ++i) col_s[lo + i] = col[buf[i]];
  } else {  // rare fallback: in-place global insertion sort
    for (int i = 1; i < n; ++i) {
      int key = idx_s[lo + i];
      int j = i - 1;
      while (j >= 0 && idx_s[lo + j] > key) { idx_s[lo + j + 1] = idx_s[lo + j]; --j; }
      idx_s[lo + j + 1] = key;
    }
    for (int i = 0; i < n; ++i) col_s[lo + i] = col[idx_s[lo + i]];
  }
}

// ---- random walk: one thread per walk; 2-load dependent chain per step ----
__global__ void walk_kernel(const float* __restrict__ rand_u,
                            const int2* __restrict__ pd,  // {rowptr, deg}
                            const int* __restrict__ col_s,
                            int* __restrict__ out_i,  // base of [2, Eout] output
                            int S, int nN, int nE, int Eout) {
  int s = blockIdx.x * blockDim.x + threadIdx.x;
  if (s >= S) return;
  const int start = s % nN;  // start = tile(arange(N), WALKS_PER_NODE)
  int cur = start;
  int* __restrict__ out_row = out_i + nE;         // added roots section
  int* __restrict__ out_col = out_i + Eout + nE;  // added targets section
  // coalesced 2x float4 load of this walk's 8 uniforms (32B aligned)
  float4 r0 = ((const float4*)rand_u)[s * 2 + 0];
  float4 r1 = ((const float4*)rand_u)[s * 2 + 1];
  float ur[WALK_LEN] = {r0.x, r0.y, r0.z, r0.w, r1.x, r1.y, r1.z, r1.w};
#pragma unroll
  for (int l = 0; l < WALK_LEN; ++l) {
    int2 c = pd[cur];                    // one b64: {rowptr, deg}
    int d = c.y;
    int off = (int)(ur[l] * (float)d);   // truncate toward zero == astype(int32)
    int dm1 = d - 1; if (dm1 < 0) dm1 = 0;
    if (off > dm1) off = dm1;
    int nxt = (d > 0) ? col_s[c.x + off] : cur;
    if (l >= 1) {
      int o = s * (WALK_LEN - 1) + (l - 1);
      out_row[o] = start;
      out_col[o] = nxt;
    }
    cur = nxt;
  }
}

// ---- bulk copy via CDNA5 async DMA: 4x16B in flight per lane --------------
#define CP_BATCH 4
#define CP_THREADS 256
__global__ void async_copy_b128_kernel(const uint8_t* __restrict__ src,
                                       uint8_t* __restrict__ dst, int nchunks) {
  __shared__ uint8_t tile[CP_THREADS * 16 * CP_BATCH];
  const int tid = threadIdx.x;
  const unsigned ldsbase = (unsigned)(unsigned long long)tile;
  const int per_block = CP_THREADS * CP_BATCH;
  for (int blockBase = blockIdx.x * per_block; blockBase < nchunks;
       blockBase += gridDim.x * per_block) {
#pragma unroll
    for (int k = 0; k < CP_BATCH; ++k) {
      int c = blockBase + k * CP_THREADS + tid;
      if (c < nchunks) {
        unsigned lds = ldsbase + (unsigned)(k * CP_THREADS + tid) * 16u;
        unsigned off = (unsigned)c * 16u;
        asm volatile("global_load_async_to_lds_b128 %0, %1, %2 offset:0"
                     :: "v"(lds), "v"(off), "s"(src) : "memory");
      }
    }
    asm volatile("s_wait_asynccnt 0" ::: "memory");  // loads done -> LDS valid
#pragma unroll
    for (int k = 0; k < CP_BATCH; ++k) {
      int c = blockBase + k * CP_THREADS + tid;
      if (c < nchunks) {
        unsigned lds = ldsbase + (unsigned)(k * CP_THREADS + tid) * 16u;
        unsigned off = (unsigned)c * 16u;
        asm volatile("global_store_async_from_lds_b128 %0, %1, %2 offset:0"
                     :: "v"(off), "v"(lds), "s"(dst) : "memory");
      }
    }
    asm volatile("s_wait_asynccnt 0" ::: "memory");  // LDS reusable
  }
}

// plain tail copy (only if byte count not 16B-divisible)
__global__ void copy_i32_kernel(const int* __restrict__ src, int* __restrict__ dst,
                                int n) {
  int i = blockIdx.x * blockDim.x + threadIdx.x;
  if (i < n) dst[i] = src[i];
}

// ---------------------------------------------------------------------------
extern "C" void kernel_launch(void* const* d_in, const int* in_sizes, int n_in,
                              void* d_out, int out_size, void* d_ws, size_t ws_size,
                              hipStream_t stream) {
  const int* edge_index = (const int*)d_in[0];     // [2, E] row-major
  const float* edge_weight = (const float*)d_in[1];
  const float* rand_unif = (const float*)d_in[2];  // [S, 8]

  const int E = in_sizes[0] / 2;
  const int S = in_sizes[2] / WALK_LEN;
  const int N = S / WALKS_PER_NODE;
  const int Eadd = S * (WALK_LEN - 1);
  const int Eout = E + Eadd;

  // workspace layout (pd first for 8B alignment)
  int2* pd = (int2*)d_ws;          // N pairs {rowptr, deg}
  int* deg = (int*)(pd + N);       // N
  int* cnt = deg + N;              // N
  int* rowptr = cnt + N;           // N+1
  int* idx_s = rowptr + (N + 1);   // E
  int* col_s = idx_s + E;          // E

  int* out_i = (int*)d_out;
  float* out_f = (float*)d_out;

  const int* row = edge_index;
  const int* col = edge_index + E;

  // 1) zero deg + cnt (contiguous 2N ints, 16B-aligned start)
  zero_i32_kernel<<<(2 * N + 1023) / 1024, 256, 0, stream>>>(deg, 2 * N);
  // 2) degree histogram
  hist_kernel<<<1024, 256, 0, stream>>>(row, deg, E);
  // 3) rowptr = exclusive scan(deg); also pack {rowptr,deg} pairs
  scan_kernel<<<1, SCAN_T, 0, stream>>>(deg, rowptr, pd, N);
  // 4) scatter edge ids into CSR slots
  scatter_kernel<<<1024, 256, 0, stream>>>(row, rowptr, cnt, idx_s, E);
  // 5) restore stable order per segment, gather sorted cols
  segsort_kernel<<<(N + 255) / 256, 256, 0, stream>>>(col, rowptr, deg, idx_s,
                                                      col_s, N);
  // 6) random walks -> added edges
  walk_kernel<<<(S + 255) / 256, 256, 0, stream>>>(rand_unif, pd, col_s, out_i,
                                                   S, N, E, Eout);
  // 7) prefix copies via batched async DMA (16B chunks; E*4 bytes per region)
  const int nchunks = E / 4;
  const int cb = (nchunks + CP_THREADS * CP_BATCH - 1) / (CP_THREADS * CP_BATCH);
  async_copy_b128_kernel<<<cb, CP_THREADS, 0, stream>>>((const uint8_t*)row,
                                                        (uint8_t*)out_i, nchunks);
  async_copy_b128_kernel<<<cb, CP_THREADS, 0, stream>>>(
      (const uint8_t*)col, (uint8_t*)(out_i + Eout), nchunks);
  async_copy_b128_kernel<<<cb, CP_THREADS, 0, stream>>>(
      (const uint8_t*)edge_weight, (uint8_t*)(out_f + 2 * Eout), nchunks);
  const int tail = E - nchunks * 4;
  if (tail > 0) {
    copy_i32_kernel<<<1, 32, 0, stream>>>(row + nchunks * 4, out_i + nchunks * 4,
                                          tail);
    copy_i32_kernel<<<1, 32, 0, stream>>>(col + nchunks * 4,
                                          out_i + Eout + nchunks * 4, tail);
    copy_i32_kernel<<<1, 32, 0, stream>>>((const int*)edge_weight + nchunks * 4,
                                          out_i + 2 * Eout + nchunks * 4, tail);
  }
  // 8) weights of added edges = 1.0f
  fill_ones_f32_kernel<<<(Eadd + 1023) / 1024, 256, 0, stream>>>(
      out_f + 2 * Eout + E, Eadd);
}